// MemoryRamTwoStreamModule_40389872452049
// MI455X (gfx1250) — compile-verified
//
#include <hip/hip_runtime.h>
#include <hip/hip_bf16.h>
#include <math.h>

#define HS     512
#define IN_DIM 2048
#define MB     100

typedef __attribute__((ext_vector_type(2))) float v2f;
typedef __attribute__((ext_vector_type(8))) float v8f;

// ---------------------------------------------------------------------------
// Phase 1: C[T x N] = X[T x K] @ W[K x N] + bias  via V_WMMA_F32_16X16X4_F32
// ---------------------------------------------------------------------------
template <bool FULL>
__global__ __launch_bounds__(256)
void proj_gemm_kernel(const float* __restrict__ X,
                      const float* __restrict__ W,     // ld == N, pre-offset rows
                      const float* __restrict__ bias,  // [N]
                      float* __restrict__ C,           // [T x N]
                      int Trows, int K, int N)
{
    __shared__ float smx[16][66];     // 16 x 64 X tile, padded vs bank conflicts

    const int tid  = threadIdx.x;
    const int lane = tid & 31;
    const int wave = tid >> 5;                         // 0..7
    const int row0 = blockIdx.y * 16;
    const int col0 = (blockIdx.x * 8 + wave) * 16;

    const int m  = lane & 15;
    const int kb = (lane >> 4) * 2;

    const int  nloc   = col0 + (lane & 15);
    const bool nvalid = FULL ? true : (nloc < N);
    const int  nc     = FULL ? nloc : (nvalid ? nloc : 0);
    const float nmask = nvalid ? 1.0f : 0.0f;

    const int lr = tid >> 4;
    const int lc = (tid & 15) * 4;
    int lrow = row0 + lr;
    if (lrow >= Trows) lrow = Trows - 1;

    v8f acc = {};

    for (int k0 = 0; k0 < K; k0 += 64) {
        const float4 xv = *(const float4*)(X + (size_t)lrow * K + k0 + lc);
        __syncthreads();
        smx[lr][lc + 0] = xv.x;
        smx[lr][lc + 1] = xv.y;
        smx[lr][lc + 2] = xv.z;
        smx[lr][lc + 3] = xv.w;
        __syncthreads();

        #pragma unroll
        for (int kk = 0; kk < 64; kk += 4) {
            v2f a, b;
            a.x = smx[m][kk + kb];
            a.y = smx[m][kk + kb + 1];
            const size_t wr = (size_t)(k0 + kk + kb) * N;
            if (FULL) {
                b.x = W[wr + nc];
                b.y = W[wr + N + nc];
            } else {
                b.x = W[wr + nc] * nmask;
                b.y = W[wr + N + nc] * nmask;
            }
            acc = __builtin_amdgcn_wmma_f32_16x16x4_f32(
                      false, a, false, b, (short)0, acc, false, false);
        }
    }

    const int crow = row0 + ((lane >> 4) ? 8 : 0);
    const int ccol = col0 + (lane & 15);
    if (FULL || ccol < N) {
        const float bv = bias[ccol];
        #pragma unroll
        for (int j = 0; j < 8; ++j) {
            const int r = crow + j;
            if (r < Trows) C[(size_t)r * N + ccol] = acc[j] + bv;
        }
    }
}

// ---------------------------------------------------------------------------
// float4 helpers (explicit component ops; fmaf -> v_fmac_f32)
// ---------------------------------------------------------------------------
__device__ inline float4 f4zero() { return make_float4(0.f, 0.f, 0.f, 0.f); }
__device__ inline float4 f4add(float4 a, float4 b) {
    a.x += b.x; a.y += b.y; a.z += b.z; a.w += b.w; return a;
}
__device__ inline float4 f4fma(float s, float4 w, float4 a) {
    a.x = fmaf(s, w.x, a.x); a.y = fmaf(s, w.y, a.y);
    a.z = fmaf(s, w.z, a.z); a.w = fmaf(s, w.w, a.w); return a;
}
__device__ inline float4 f4relu(float4 a) {
    a.x = fmaxf(a.x, 0.f); a.y = fmaxf(a.y, 0.f);
    a.z = fmaxf(a.z, 0.f); a.w = fmaxf(a.w, 0.f); return a;
}

// ---------------------------------------------------------------------------
// wave-level softmax over MB=100 entries in LDS buf[0..99]; 'lane' = 0..31
// ---------------------------------------------------------------------------
__device__ inline void softmax_mb(float* buf, int lane)
{
    float v0 = buf[lane];
    float v1 = buf[lane + 32];
    float v2 = buf[lane + 64];
    float v3 = (lane + 96 < MB) ? buf[lane + 96] : -__builtin_inff();
    float mx = fmaxf(fmaxf(v0, v1), fmaxf(v2, v3));
    #pragma unroll
    for (int off = 16; off > 0; off >>= 1) mx = fmaxf(mx, __shfl_xor(mx, off, 32));
    float e0 = expf(v0 - mx), e1 = expf(v1 - mx), e2 = expf(v2 - mx);
    float e3 = (lane + 96 < MB) ? expf(v3 - mx) : 0.0f;
    float s = e0 + e1 + e2 + e3;
    #pragma unroll
    for (int off = 16; off > 0; off >>= 1) s += __shfl_xor(s, off, 32);
    const float inv = 1.0f / s;
    buf[lane]       = e0 * inv;
    buf[lane + 32]  = e1 * inv;
    buf[lane + 64]  = e2 * inv;
    if (lane + 96 < MB) buf[lane + 96] = e3 * inv;
}

// ---------------------------------------------------------------------------
// Phase 2: sequential recurrence, one persistent 1024-thread workgroup.
// mem[100x512] + all state in LDS (~236 KB of the WGP's 320 KB).
// All weight streaming uses global_load_b128 (float4) to saturate L2->WGP BW.
// ---------------------------------------------------------------------------
__global__ __launch_bounds__(1024)
void recurrent_kernel(const float* __restrict__ W_rp,  const float* __restrict__ b_rp,
                      const float* __restrict__ W_wp,  const float* __restrict__ b_wp,
                      const float* __restrict__ W_rh,  const float* __restrict__ b_rh,
                      const float* __restrict__ W_rha,                   // [3072 x 512]
                      const float* __restrict__ W_rhm,                   // [3072 x 512]
                      const float* __restrict__ W_ca,                    // [2560 x 512]
                      const float* __restrict__ W_cm,                    // [2560 x 512]
                      const float* __restrict__ W_wpa,                   // [2560 x 100]
                      const float* __restrict__ W_wpm,                   // [2560 x 100]
                      const float* __restrict__ P_rha, const float* __restrict__ P_rhm,
                      const float* __restrict__ P_ca,  const float* __restrict__ P_cm,
                      const float* __restrict__ P_wpa, const float* __restrict__ P_wpm,
                      float* __restrict__ out,         // [T x 512]
                      int nT)
{
    const int tid = threadIdx.x;

    __shared__ __attribute__((aligned(16))) float sm_mem[MB * HS];   // 200 KB
    __shared__ __attribute__((aligned(16))) float sm_xh[3 * HS];     // [ha|hm|h]
    __shared__ __attribute__((aligned(16))) float sm_r[HS];
    __shared__ __attribute__((aligned(16))) float sm_h1[HS];
    __shared__ __attribute__((aligned(16))) float sm_ha1[HS];
    __shared__ __attribute__((aligned(16))) float sm_hm1[HS];
    __shared__ __attribute__((aligned(16))) float sm_ca[HS];
    __shared__ __attribute__((aligned(16))) float sm_cm[HS];
    __shared__ __attribute__((aligned(16))) float sm_ar[128];
    __shared__ __attribute__((aligned(16))) float sm_awa[128];
    __shared__ __attribute__((aligned(16))) float sm_awm[128];
    __shared__ float sm_aw[4];
    __shared__ __attribute__((aligned(16))) float4 sm_red4[1024];    // 16 KB

    for (int i = tid; i < 3 * HS; i += 1024) sm_xh[i] = 0.0f;
    for (int i = tid; i < MB * HS; i += 1024) sm_mem[i] = 0.0f;
    __syncthreads();

    const int q128 = tid & 127;          // output quad for 512-wide stages
    const int kg8  = tid >> 7;           // 0..7 split-K group
    const int q32  = tid & 31;           // output quad for 100-wide stages (0..24 valid)
    const int ig32 = tid >> 5;           // 0..31 split-K group

    for (int t = 0; t < nT; ++t) {
        // ---- ar logits = xh @ W_rp + b_rp   (K=1536, N=100=25 quads) -------
        if (q32 < 25) {
            const int i0 = ig32 * 48;
            const float* Wp = W_rp + (size_t)i0 * MB + q32 * 4;
            float4 acc = f4zero();
            for (int ii = 0; ii < 48; ++ii)
                acc = f4fma(sm_xh[i0 + ii], *(const float4*)(Wp + (size_t)ii * MB), acc);
            sm_red4[tid] = acc;
        }
        __syncthreads();
        if (tid < 25) {
            float4 s = *(const float4*)(b_rp + tid * 4);
            for (int g = 0; g < 32; ++g) s = f4add(s, sm_red4[g * 32 + tid]);
            *(float4*)(sm_ar + tid * 4) = s;
        }
        __syncthreads();

        // ---- aw logits = xh @ W_wp + b_wp   (K=1536, N=3) ------------------
        if (tid < 96) {
            const int j = tid / 32, lane = tid & 31;
            float acc = 0.0f;
            for (int i = lane; i < 3 * HS; i += 32) acc += sm_xh[i] * W_wp[i * 3 + j];
            #pragma unroll
            for (int off = 16; off > 0; off >>= 1) acc += __shfl_xor(acc, off, 32);
            if (lane == 0) sm_aw[j] = acc + b_wp[j];
        }
        __syncthreads();

        // ---- softmax(ar) on wave0, softmax(aw) on thread 32 ----------------
        if (tid < 32) {
            softmax_mb(sm_ar, tid);
        } else if (tid == 32) {
            float a0 = sm_aw[0], a1 = sm_aw[1], a2 = sm_aw[2];
            float mx = fmaxf(a0, fmaxf(a1, a2));
            float e0 = expf(a0 - mx), e1 = expf(a1 - mx), e2 = expf(a2 - mx);
            float s  = e0 + e1 + e2;
            sm_aw[0] = e0 / s; sm_aw[1] = e1 / s; sm_aw[2] = e2 / s;
        }
        __syncthreads();

        // ---- r = ar @ mem   (100 x 512) ------------------------------------
        {
            const int js = (kg8 * MB) >> 3;
            const int je = ((kg8 + 1) * MB) >> 3;
            const float* mp = sm_mem + q128 * 4;
            float4 acc = f4zero();
            for (int j = js; j < je; ++j)
                acc = f4fma(sm_ar[j], *(const float4*)(mp + j * HS), acc);
            sm_red4[tid] = acc;
        }
        __syncthreads();
        if (tid < 128) {
            float4 s = sm_red4[tid];
            #pragma unroll
            for (int g = 1; g < 8; ++g) s = f4add(s, sm_red4[g * 128 + tid]);
            *(float4*)(sm_r + tid * 4) = s;
        }
        __syncthreads();

        // ---- h1 = relu([r,h] @ W_rh + b_rh)   (K=1024, 8x128 chunks) -------
        {
            const float* xsrc = (kg8 < 4) ? (sm_r + kg8 * 128)
                                          : (sm_xh + 2 * HS + (kg8 - 4) * 128);
            const float* Wp = W_rh + (size_t)(kg8 * 128) * HS + q128 * 4;
            float4 acc = f4zero();
            for (int ii = 0; ii < 128; ++ii)
                acc = f4fma(xsrc[ii], *(const float4*)(Wp + (size_t)ii * HS), acc);
            sm_red4[tid] = acc;
        }
        __syncthreads();
        if (tid < 128) {
            float4 s = *(const float4*)(b_rh + tid * 4);
            #pragma unroll
            for (int g = 0; g < 8; ++g) s = f4add(s, sm_red4[g * 128 + tid]);
            *(float4*)(sm_h1 + tid * 4) = f4relu(s);
        }
        __syncthreads();

        // ---- ha1 = relu(P_rha[t] + [r,ha] @ W_rha[2048:,:]) ----------------
        {
            const float* xsrc = (kg8 < 4) ? (sm_r + kg8 * 128)
                                          : (sm_xh + (kg8 - 4) * 128);       // ha
            const float* Wp = W_rha + (size_t)(IN_DIM + kg8 * 128) * HS + q128 * 4;
            float4 acc = f4zero();
            for (int ii = 0; ii < 128; ++ii)
                acc = f4fma(xsrc[ii], *(const float4*)(Wp + (size_t)ii * HS), acc);
            sm_red4[tid] = acc;
        }
        __syncthreads();
        if (tid < 128) {
            float4 s = *(const float4*)(P_rha + (size_t)t * HS + tid * 4);
            #pragma unroll
            for (int g = 0; g < 8; ++g) s = f4add(s, sm_red4[g * 128 + tid]);
            *(float4*)(sm_ha1 + tid * 4) = f4relu(s);
        }
        __syncthreads();

        // ---- hm1 = relu(P_rhm[t] + [r,hm] @ W_rhm[2048:,:]) ----------------
        {
            const float* xsrc = (kg8 < 4) ? (sm_r + kg8 * 128)
                                          : (sm_xh + HS + (kg8 - 4) * 128);  // hm
            const float* Wp = W_rhm + (size_t)(IN_DIM + kg8 * 128) * HS + q128 * 4;
            float4 acc = f4zero();
            for (int ii = 0; ii < 128; ++ii)
                acc = f4fma(xsrc[ii], *(const float4*)(Wp + (size_t)ii * HS), acc);
            sm_red4[tid] = acc;
        }
        __syncthreads();
        if (tid < 128) {
            float4 s = *(const float4*)(P_rhm + (size_t)t * HS + tid * 4);
            #pragma unroll
            for (int g = 0; g < 8; ++g) s = f4add(s, sm_red4[g * 128 + tid]);
            *(float4*)(sm_hm1 + tid * 4) = f4relu(s);
        }
        __syncthreads();

        // ---- c_a = relu(P_ca[t] + ha @ W_ca[0:512,:])   (K=512, 8x64) ------
        {
            const float* xsrc = sm_xh + kg8 * 64;                            // ha
            const float* Wp = W_ca + (size_t)(kg8 * 64) * HS + q128 * 4;
            float4 acc = f4zero();
            for (int ii = 0; ii < 64; ++ii)
                acc = f4fma(xsrc[ii], *(const float4*)(Wp + (size_t)ii * HS), acc);
            sm_red4[tid] = acc;
        }
        __syncthreads();
        if (tid < 128) {
            float4 s = *(const float4*)(P_ca + (size_t)t * HS + tid * 4);
            #pragma unroll
            for (int g = 0; g < 8; ++g) s = f4add(s, sm_red4[g * 128 + tid]);
            *(float4*)(sm_ca + tid * 4) = f4relu(s);
        }
        __syncthreads();

        // ---- c_m = relu(P_cm[t] + hm @ W_cm[0:512,:]) ----------------------
        {
            const float* xsrc = sm_xh + HS + kg8 * 64;                       // hm
            const float* Wp = W_cm + (size_t)(kg8 * 64) * HS + q128 * 4;
            float4 acc = f4zero();
            for (int ii = 0; ii < 64; ++ii)
                acc = f4fma(xsrc[ii], *(const float4*)(Wp + (size_t)ii * HS), acc);
            sm_red4[tid] = acc;
        }
        __syncthreads();
        if (tid < 128) {
            float4 s = *(const float4*)(P_cm + (size_t)t * HS + tid * 4);
            #pragma unroll
            for (int g = 0; g < 8; ++g) s = f4add(s, sm_red4[g * 128 + tid]);
            *(float4*)(sm_cm + tid * 4) = f4relu(s);
        }
        __syncthreads();

        // ---- aw_a logits = P_wpa[t] + ha @ W_wpa[0:512,:]  (K=512, 25 quads)
        if (q32 < 25) {
            const int i0 = ig32 * 16;
            const float* Wp = W_wpa + (size_t)i0 * MB + q32 * 4;
            float4 acc = f4zero();
            for (int ii = 0; ii < 16; ++ii)
                acc = f4fma(sm_xh[i0 + ii], *(const float4*)(Wp + (size_t)ii * MB), acc);
            sm_red4[tid] = acc;
        }
        __syncthreads();
        if (tid < 25) {
            float4 s = *(const float4*)(P_wpa + (size_t)t * MB + tid * 4);
            for (int g = 0; g < 32; ++g) s = f4add(s, sm_red4[g * 32 + tid]);
            *(float4*)(sm_awa + tid * 4) = s;
        }
        __syncthreads();

        // ---- aw_m logits = P_wpm[t] + hm @ W_wpm[0:512,:] ------------------
        if (q32 < 25) {
            const int i0 = ig32 * 16;
            const float* Wp = W_wpm + (size_t)i0 * MB + q32 * 4;
            float4 acc = f4zero();
            for (int ii = 0; ii < 16; ++ii)
                acc = f4fma(sm_xh[HS + i0 + ii], *(const float4*)(Wp + (size_t)ii * MB), acc);
            sm_red4[tid] = acc;
        }
        __syncthreads();
        if (tid < 25) {
            float4 s = *(const float4*)(P_wpm + (size_t)t * MB + tid * 4);
            for (int g = 0; g < 32; ++g) s = f4add(s, sm_red4[g * 32 + tid]);
            *(float4*)(sm_awm + tid * 4) = s;
        }
        __syncthreads();

        // ---- dual softmax: aw_a on wave0, aw_m on wave1 --------------------
        if (tid < 32)       softmax_mb(sm_awa, tid);
        else if (tid < 64)  softmax_mb(sm_awm, tid - 32);
        __syncthreads();

        // ---- mem update + output + state rotation --------------------------
        {
            const float aw0 = sm_aw[0], aw1 = sm_aw[1], aw2 = sm_aw[2];
            for (int qd = tid; qd < (MB * HS) / 4; qd += 1024) {
                const int j  = qd >> 7;                 // element/512
                const int n4 = (qd & 127) * 4;
                float4 mv = *(const float4*)(sm_mem + qd * 4);
                const float4 cav = *(const float4*)(sm_ca + n4);
                const float4 cmv = *(const float4*)(sm_cm + n4);
                const float wa = aw1 * sm_awa[j];
                const float wm = aw2 * sm_awm[j];
                mv.x = fmaf(aw0, mv.x, fmaf(wa, cav.x, wm * cmv.x));
                mv.y = fmaf(aw0, mv.y, fmaf(wa, cav.y, wm * cmv.y));
                mv.z = fmaf(aw0, mv.z, fmaf(wa, cav.z, wm * cmv.z));
                mv.w = fmaf(aw0, mv.w, fmaf(wa, cav.w, wm * cmv.w));
                *(float4*)(sm_mem + qd * 4) = mv;
            }
            if (tid < 128) {
                const float4 h1v = *(const float4*)(sm_h1 + tid * 4);
                *(float4*)(out + (size_t)t * HS + tid * 4) = h1v;
                *(float4*)(sm_xh + tid * 4)           = *(const float4*)(sm_ha1 + tid * 4);
                *(float4*)(sm_xh + HS + tid * 4)      = *(const float4*)(sm_hm1 + tid * 4);
                *(float4*)(sm_xh + 2 * HS + tid * 4)  = h1v;
            }
            __syncthreads();
        }
    }
}

// ---------------------------------------------------------------------------
extern "C" void kernel_launch(void* const* d_in, const int* in_sizes, int n_in,
                              void* d_out, int out_size, void* d_ws, size_t ws_size,
                              hipStream_t stream)
{
    (void)n_in; (void)out_size; (void)ws_size;

    const float* Xa    = (const float*)d_in[0];
    const float* Xm    = (const float*)d_in[1];
    const float* W_ca  = (const float*)d_in[2];  const float* b_ca  = (const float*)d_in[3];
    const float* W_cm  = (const float*)d_in[4];  const float* b_cm  = (const float*)d_in[5];
    const float* W_wp  = (const float*)d_in[6];  const float* b_wp  = (const float*)d_in[7];
    const float* W_wpa = (const float*)d_in[8];  const float* b_wpa = (const float*)d_in[9];
    const float* W_wpm = (const float*)d_in[10]; const float* b_wpm = (const float*)d_in[11];
    const float* W_rp  = (const float*)d_in[12]; const float* b_rp  = (const float*)d_in[13];
    const float* W_rh  = (const float*)d_in[14]; const float* b_rh  = (const float*)d_in[15];
    const float* W_rha = (const float*)d_in[16]; const float* b_rha = (const float*)d_in[17];
    const float* W_rhm = (const float*)d_in[18]; const float* b_rhm = (const float*)d_in[19];

    const int T = in_sizes[0] / IN_DIM;        // nImg == T in this harness

    float* ws = (float*)d_ws;
    size_t off = 0;
    float* P_rha = ws + off; off += (size_t)T * HS;
    float* P_rhm = ws + off; off += (size_t)T * HS;
    float* P_ca  = ws + off; off += (size_t)T * HS;
    float* P_cm  = ws + off; off += (size_t)T * HS;
    float* P_wpa = ws + off; off += (size_t)T * MB;
    float* P_wpm = ws + off; off += (size_t)T * MB;

    const dim3 blk(256, 1, 1);
    const dim3 gHS((HS + 127) / 128, (T + 15) / 16, 1);
    const dim3 gMB((MB + 127) / 128, (T + 15) / 16, 1);

    // Phase 1: hoisted input projections (WMMA f32).
    proj_gemm_kernel<true ><<<gHS, blk, 0, stream>>>(Xa, W_rha,                   b_rha, P_rha, T, IN_DIM, HS);
    proj_gemm_kernel<true ><<<gHS, blk, 0, stream>>>(Xm, W_rhm,                   b_rhm, P_rhm, T, IN_DIM, HS);
    proj_gemm_kernel<true ><<<gHS, blk, 0, stream>>>(Xa, W_ca  + (size_t)HS * HS, b_ca,  P_ca,  T, IN_DIM, HS);
    proj_gemm_kernel<true ><<<gHS, blk, 0, stream>>>(Xm, W_cm  + (size_t)HS * HS, b_cm,  P_cm,  T, IN_DIM, HS);
    proj_gemm_kernel<false><<<gMB, blk, 0, stream>>>(Xa, W_wpa + (size_t)HS * MB, b_wpa, P_wpa, T, IN_DIM, MB);
    proj_gemm_kernel<false><<<gMB, blk, 0, stream>>>(Xm, W_wpm + (size_t)HS * MB, b_wpm, P_wpm, T, IN_DIM, MB);

    // Phase 2: sequential recurrence (mem + state fully LDS-resident,
    // float4 weight streaming)
    recurrent_kernel<<<dim3(1), dim3(1024), 0, stream>>>(
        W_rp, b_rp, W_wp, b_wp, W_rh, b_rh,
        W_rha, W_rhm, W_ca, W_cm, W_wpa, W_wpm,
        P_rha, P_rhm, P_ca, P_cm, P_wpa, P_wpm,
        (float*)d_out, T);
}